// GCN_74225624809997
// MI455X (gfx1250) — compile-verified
//
#include <hip/hip_runtime.h>
#include <hip/hip_bf16.h>
#include <math.h>

typedef __attribute__((ext_vector_type(2))) float v2f;
typedef __attribute__((ext_vector_type(8))) float v8f;

// ---------------------------------------------------------------------------
// Utility kernels
// ---------------------------------------------------------------------------
__global__ void gcn_zero_f32(float* __restrict__ p, long long n) {
    long long t = (long long)blockIdx.x * blockDim.x + threadIdx.x;
    if (t < n) p[t] = 0.0f;
}

__global__ void gcn_count_deg(const int* __restrict__ dst, float* __restrict__ deg, int E) {
    int e = blockIdx.x * blockDim.x + threadIdx.x;
    if (e < E) atomicAdd(&deg[dst[e]], 1.0f);
}

__global__ void gcn_dinv(float* __restrict__ deg_io, int n) {
    int i = blockIdx.x * blockDim.x + threadIdx.x;
    if (i < n) deg_io[i] = rsqrtf(deg_io[i] + 1.0f);  // +1 self loop
}

// agg = h * dinv(i)^2 + bias[d]   (self-loop contribution + bias, init for scatter)
__global__ void gcn_selfloop_bias(const float* __restrict__ h, const float* __restrict__ dinv,
                                  const float* __restrict__ bias, float* __restrict__ agg,
                                  int Nn, int D) {
    long long t = (long long)blockIdx.x * blockDim.x + threadIdx.x;
    long long total = (long long)Nn * D;
    if (t >= total) return;
    int i = (int)(t / D);
    int d = (int)(t % D);
    float di = dinv[i];
    agg[t] = h[t] * di * di + bias[d];
}

// scatter-add over edges: agg[dst] += h[src] * dinv[src]*dinv[dst]
// thread layout: D/4 consecutive threads per edge (one wave per edge at D=128)
__global__ void gcn_scatter_edges(const float* __restrict__ h, const float* __restrict__ dinv,
                                  const int* __restrict__ src, const int* __restrict__ dst,
                                  float* __restrict__ agg, int E, int D) {
    long long t = (long long)blockIdx.x * blockDim.x + threadIdx.x;
    int per = D >> 2;
    long long total = (long long)E * per;
    if (t >= total) return;
    int e  = (int)(t / per);
    int c4 = (int)(t % per) << 2;
    int s = src[e], d = dst[e];
    float coef = dinv[s] * dinv[d];
    const float4 hv = *(const float4*)(&h[(long long)s * D + c4]);
    float* base = &agg[(long long)d * D + c4];
    atomicAdd(base + 0, hv.x * coef);
    atomicAdd(base + 1, hv.y * coef);
    atomicAdd(base + 2, hv.z * coef);
    atomicAdd(base + 3, hv.w * coef);
}

__global__ void gcn_relu(float* __restrict__ p, long long n) {
    long long t = (long long)blockIdx.x * blockDim.x + threadIdx.x;
    if (t < n) p[t] = fmaxf(p[t], 0.0f);
}

// ---------------------------------------------------------------------------
// FP32 GEMM via V_WMMA_F32_16X16X4_F32 (exact fp32 FMA semantics).
// One wave computes one 16x16 tile of C; 4 waves/block -> 64 columns/block.
// Requires M % 16 == 0 and N % 64 == 0 or N == 64 exactly (no partial waves,
// EXEC all-1s as WMMA requires). A: MxK row-major, B: KxN row-major.
//
// A operand (16x4 f32, ISA 7.12.2): lanes 0-15 hold row M=lane K={0,1} in
// VGPR0/1, lanes 16-31 hold K={2,3}. B mirrored over columns. C/D: VGPR v
// holds row v (lanes 0-15) / row v+8 (lanes 16-31), col = lane&15.
// ---------------------------------------------------------------------------
__global__ void gcn_gemm_f32_wmma(const float* __restrict__ A, const float* __restrict__ B,
                                  float* __restrict__ C, int M, int N, int K) {
    const int lane = threadIdx.x & 31;
    const int wave = threadIdx.x >> 5;
    const int m0 = blockIdx.x * 16;
    const int n0 = blockIdx.y * 64 + wave * 16;
    if (m0 >= M || n0 >= N) return;  // wave-uniform exit
    const int half = lane >> 4;   // 0: K pair {0,1}, 1: K pair {2,3}
    const int l15  = lane & 15;
    const long long arow = (long long)(m0 + l15) * K;
    const int n = n0 + l15;

    v8f c = {0.f, 0.f, 0.f, 0.f, 0.f, 0.f, 0.f, 0.f};
    for (int k = 0; k < K; k += 4) {
        const int ka = k + (half << 1);
        v2f a, b;
        a.x = A[arow + ka];
        a.y = A[arow + ka + 1];
        b.x = B[(long long)ka * N + n];
        b.y = B[(long long)(ka + 1) * N + n];
        c = __builtin_amdgcn_wmma_f32_16x16x4_f32(
                /*neg_a=*/false, a, /*neg_b=*/false, b,
                /*c_mod=*/(short)0, c, /*reuse_a=*/false, /*reuse_b=*/false);
    }
    const int rbase = m0 + half * 8;
#pragma unroll
    for (int v = 0; v < 8; ++v) {
        C[(long long)(rbase + v) * N + n] = c[v];
    }
}

// ---------------------------------------------------------------------------
// Output head: logits = h @ Wo + bo (64x2), then log_softmax over 2 classes.
// ---------------------------------------------------------------------------
__global__ void gcn_out_head(const float* __restrict__ h, const float* __restrict__ Wo,
                             const float* __restrict__ bo, float* __restrict__ out, int Nn) {
    int i = blockIdx.x * blockDim.x + threadIdx.x;
    if (i >= Nn) return;
    const float* hr = h + (long long)i * 64;
    float l0 = bo[0], l1 = bo[1];
#pragma unroll 8
    for (int k = 0; k < 64; ++k) {
        float hv = hr[k];
        l0 = fmaf(hv, Wo[k * 2 + 0], l0);
        l1 = fmaf(hv, Wo[k * 2 + 1], l1);
    }
    float m = fmaxf(l0, l1);
    float lse = m + logf(expf(l0 - m) + expf(l1 - m));
    out[(long long)i * 2 + 0] = l0 - lse;
    out[(long long)i * 2 + 1] = l1 - lse;
}

// ---------------------------------------------------------------------------
// Host-side launch sequence (all kernels on `stream`, in order).
// ---------------------------------------------------------------------------
static inline long long cdiv_ll(long long a, long long b) { return (a + b - 1) / b; }

extern "C" void kernel_launch(void* const* d_in, const int* in_sizes, int n_in,
                              void* d_out, int out_size, void* d_ws, size_t ws_size,
                              hipStream_t stream) {
    const float* x    = (const float*)d_in[0];
    const int*   ei   = (const int*)  d_in[1];
    const float* W1   = (const float*)d_in[2];
    const float* b1   = (const float*)d_in[3];
    const float* W2   = (const float*)d_in[4];
    const float* b2   = (const float*)d_in[5];
    const float* Wo   = (const float*)d_in[6];
    const float* bo   = (const float*)d_in[7];
    float*       out  = (float*)d_out;

    const int D_IN = 128, H1 = 128, H2 = 64;
    const int Nn = in_sizes[0] / D_IN;     // 50000
    const int E  = in_sizes[1] / 2;        // 800000
    const int* src = ei;
    const int* dst = ei + E;

    // Workspace layout (bytes, 256-aligned):
    //   dinv : Nn floats
    //   hbuf : Nn * 128 floats   (h1, later h2)
    //   abuf : Nn * 128 floats   (agg1, later agg2)
    char* ws = (char*)d_ws;
    size_t off = 0;
    float* dinv = (float*)(ws + off); off += ((size_t)Nn * 4 + 255) & ~(size_t)255;
    float* hbuf = (float*)(ws + off); off += ((size_t)Nn * D_IN * 4 + 255) & ~(size_t)255;
    float* abuf = (float*)(ws + off);

    const int BT = 256;

    // ---- degree / normalization -------------------------------------------
    gcn_zero_f32<<<(unsigned)cdiv_ll(Nn, BT), BT, 0, stream>>>(dinv, Nn);
    gcn_count_deg<<<(unsigned)cdiv_ll(E, BT), BT, 0, stream>>>(dst, dinv, E);
    gcn_dinv<<<(unsigned)cdiv_ll(Nn, BT), BT, 0, stream>>>(dinv, Nn);

    // ---- layer 1: h1 = x @ W1 ; agg1 = norm-aggregate(h1) + b1 ; relu -----
    {
        dim3 grid(Nn / 16, H1 / 64);
        gcn_gemm_f32_wmma<<<grid, 128, 0, stream>>>(x, W1, hbuf, Nn, H1, D_IN);
    }
    {
        long long tot = (long long)Nn * H1;
        gcn_selfloop_bias<<<(unsigned)cdiv_ll(tot, BT), BT, 0, stream>>>(hbuf, dinv, b1, abuf, Nn, H1);
        long long etot = (long long)E * (H1 / 4);
        gcn_scatter_edges<<<(unsigned)cdiv_ll(etot, BT), BT, 0, stream>>>(hbuf, dinv, src, dst, abuf, E, H1);
        gcn_relu<<<(unsigned)cdiv_ll(tot, BT), BT, 0, stream>>>(abuf, tot);
    }

    // ---- layer 2: h2 = agg1 @ W2 ; agg2 = norm-aggregate(h2) + b2 ; relu --
    {
        dim3 grid(Nn / 16, H2 / 64);
        gcn_gemm_f32_wmma<<<grid, 128, 0, stream>>>(abuf, W2, hbuf, Nn, H2, H1);
    }
    {
        long long tot = (long long)Nn * H2;
        // agg2 reuses abuf (h2 lives in hbuf; abuf contents fully overwritten)
        gcn_selfloop_bias<<<(unsigned)cdiv_ll(tot, BT), BT, 0, stream>>>(hbuf, dinv, b2, abuf, Nn, H2);
        long long etot = (long long)E * (H2 / 4);
        gcn_scatter_edges<<<(unsigned)cdiv_ll(etot, BT), BT, 0, stream>>>(hbuf, dinv, src, dst, abuf, E, H2);
        gcn_relu<<<(unsigned)cdiv_ll(tot, BT), BT, 0, stream>>>(abuf, tot);
    }

    // ---- output head: log_softmax(agg2 @ Wo + bo) -------------------------
    gcn_out_head<<<(unsigned)cdiv_ll(Nn, BT), BT, 0, stream>>>(abuf, Wo, bo, out, Nn);
}